// Coords2RMSD_75874892251827
// MI455X (gfx1250) — compile-verified
//
#include <hip/hip_runtime.h>
#include <hip/hip_bf16.h>
#include <math.h>

// ---------------------------------------------------------------------------
// Kabsch RMSD, single-pass raw-moment formulation.
//   Per structure b:
//     Sx,Sy (3 each), Sxx=sum|x|^2, Syy=sum|y|^2, M = sum x y^T  (masked)
//     C = M - Sx Sy^T / n
//     E = (Sxx - |Sx|^2/n) + (Syy - |Sy|^2/n)
//     G = C^T C  (computed with V_WMMA_F32_16X16X4_F32, zero-padded 3x3)
//     eig(G) closed-form (symmetric 3x3, trig method) -> singular values
//     rmsd = sqrt(max(E - 2*(s0+s1+sign(det C)*s2), 0)/n)
// Memory-bound: data read once -> non-temporal B128 loads, no LDS staging.
// ---------------------------------------------------------------------------

typedef __attribute__((ext_vector_type(2))) float v2f;
typedef __attribute__((ext_vector_type(4))) float v4f;
typedef __attribute__((ext_vector_type(8))) float v8f;

#define THREADS 256
#define NWAVES  (THREADS / 32)

__global__ __launch_bounds__(THREADS)
void kabsch_rmsd_kernel(const float* __restrict__ x,
                        const float* __restrict__ y,
                        const int*   __restrict__ num_atoms,
                        float*       __restrict__ out,
                        int N)
{
    const int b    = blockIdx.x;
    const int tid  = threadIdx.x;
    const int lane = tid & 31;
    const int wid  = tid >> 5;

    const int   n_i = num_atoms[b];
    const float nf  = (float)n_i;

    const long long base = (long long)b * 3LL * (long long)N;
    const v4f* __restrict__ px = (const v4f*)(x + base);
    const v4f* __restrict__ py = (const v4f*)(y + base);

    // acc[0..2]=Sx  acc[3..5]=Sy  acc[6]=Sxx  acc[7]=Syy  acc[8..16]=M row-major
    float acc[17];
#pragma unroll
    for (int k = 0; k < 17; ++k) acc[k] = 0.0f;

    // Each thread handles 4 atoms (48 B per array) per iteration:
    // wave covers 1536 contiguous bytes -> fully coalesced B128 NT loads.
    for (int a0 = tid * 4; a0 < N; a0 += THREADS * 4) {
        float X[4][3], Y[4][3];
        if (a0 + 4 <= N) {
            const int k = (a0 >> 2) * 3;   // float4 index: 12 floats / 4
            v4f q0 = __builtin_nontemporal_load(px + k + 0);
            v4f q1 = __builtin_nontemporal_load(px + k + 1);
            v4f q2 = __builtin_nontemporal_load(px + k + 2);
            v4f r0 = __builtin_nontemporal_load(py + k + 0);
            v4f r1 = __builtin_nontemporal_load(py + k + 1);
            v4f r2 = __builtin_nontemporal_load(py + k + 2);
            X[0][0]=q0.x; X[0][1]=q0.y; X[0][2]=q0.z;
            X[1][0]=q0.w; X[1][1]=q1.x; X[1][2]=q1.y;
            X[2][0]=q1.z; X[2][1]=q1.w; X[2][2]=q2.x;
            X[3][0]=q2.y; X[3][1]=q2.z; X[3][2]=q2.w;
            Y[0][0]=r0.x; Y[0][1]=r0.y; Y[0][2]=r0.z;
            Y[1][0]=r0.w; Y[1][1]=r1.x; Y[1][2]=r1.y;
            Y[2][0]=r1.z; Y[2][1]=r1.w; Y[2][2]=r2.x;
            Y[3][0]=r2.y; Y[3][1]=r2.z; Y[3][2]=r2.w;
        } else {
            // generic tail (not hit for N=8192)
#pragma unroll
            for (int j = 0; j < 4; ++j) {
                const int a = a0 + j;
                const bool ok = (a < N);
                const long long e = 3LL * a;
#pragma unroll
                for (int c = 0; c < 3; ++c) {
                    X[j][c] = ok ? x[base + e + c] : 0.0f;
                    Y[j][c] = ok ? y[base + e + c] : 0.0f;
                }
            }
        }
#pragma unroll
        for (int j = 0; j < 4; ++j) {
            const float w  = ((a0 + j) < n_i) ? 1.0f : 0.0f;
            const float x0 = X[j][0] * w, x1 = X[j][1] * w, x2 = X[j][2] * w;
            const float y0 = Y[j][0] * w, y1 = Y[j][1] * w, y2 = Y[j][2] * w;
            acc[0] += x0;  acc[1] += x1;  acc[2] += x2;
            acc[3] += y0;  acc[4] += y1;  acc[5] += y2;
            acc[6] = fmaf(x0, x0, fmaf(x1, x1, fmaf(x2, x2, acc[6])));
            acc[7] = fmaf(y0, y0, fmaf(y1, y1, fmaf(y2, y2, acc[7])));
            acc[8]  = fmaf(x0, y0, acc[8]);   acc[9]  = fmaf(x0, y1, acc[9]);
            acc[10] = fmaf(x0, y2, acc[10]);  acc[11] = fmaf(x1, y0, acc[11]);
            acc[12] = fmaf(x1, y1, acc[12]);  acc[13] = fmaf(x1, y2, acc[13]);
            acc[14] = fmaf(x2, y0, acc[14]);  acc[15] = fmaf(x2, y1, acc[15]);
            acc[16] = fmaf(x2, y2, acc[16]);
        }
    }

    // ---- wave32 butterfly reduction, then cross-wave via LDS --------------
    __shared__ float red[NWAVES][20];
#pragma unroll
    for (int k = 0; k < 17; ++k) {
        float v = acc[k];
#pragma unroll
        for (int off = 16; off > 0; off >>= 1)
            v += __shfl_xor(v, off, 32);
        if (lane == 0) red[wid][k] = v;
    }
    __syncthreads();

    if (tid < 32) {                       // wave 0 only, all 32 lanes active
        float tot = 0.0f;
        if (lane < 17) {
#pragma unroll
            for (int w = 0; w < NWAVES; ++w) tot += red[w][lane];
        }
        // broadcast all 17 sums to every lane of the wave
        float s[17];
#pragma unroll
        for (int i = 0; i < 17; ++i) s[i] = __shfl(tot, i, 32);

        const float inv_n = 1.0f / nf;
        float Cm[3][3];
#pragma unroll
        for (int i = 0; i < 3; ++i)
#pragma unroll
            for (int j = 0; j < 3; ++j)
                Cm[i][j] = s[8 + 3 * i + j] - s[i] * s[3 + j] * inv_n;

        const float E =
            (s[6] - (s[0]*s[0] + s[1]*s[1] + s[2]*s[2]) * inv_n) +
            (s[7] - (s[3]*s[3] + s[4]*s[4] + s[5]*s[5]) * inv_n);

        const float detC =
            Cm[0][0] * (Cm[1][1]*Cm[2][2] - Cm[1][2]*Cm[2][1]) -
            Cm[0][1] * (Cm[1][0]*Cm[2][2] - Cm[1][2]*Cm[2][0]) +
            Cm[0][2] * (Cm[1][0]*Cm[2][1] - Cm[1][1]*Cm[2][0]);

        // ---- G = C^T C via V_WMMA_F32_16X16X4_F32 (zero-padded) ----------
        // A (16x4, MxK): A[m][k] = C[k][m].  Layout: lanes 0-15 row m hold
        // {K0,K1}; lanes 16-31 row m hold {K2,K3}.
        // B (4x16, KxN): B[k][n] = C[k][n].  VGPR0: K0 (lanes0-15)/K1
        // (lanes16-31); VGPR1: K2 (lanes0-15)/K3 (lanes16-31).
        v2f Aop = (v2f)(0.0f), Bop = (v2f)(0.0f);
        if (lane < 3) {
            Aop.x = Cm[0][lane];  Aop.y = Cm[1][lane];   // K=0,1 of column 'lane'
            Bop.x = Cm[0][lane];  Bop.y = Cm[2][lane];   // rows K=0 and K=2
        } else if (lane >= 16 && lane < 19) {
            const int m = lane - 16;
            Aop.x = Cm[2][m];                            // K=2 (K=3 pad = 0)
            Bop.x = Cm[1][m];                            // row K=1 (K=3 pad = 0)
        }
        v8f Cz = {};
        v8f G = __builtin_amdgcn_wmma_f32_16x16x4_f32(
            /*neg_a=*/false, Aop, /*neg_b=*/false, Bop,
            /*c_mod=*/(short)0, Cz, /*reuse_a=*/false, /*reuse_b=*/false);

        // D layout: VGPR r, lanes 0-15 hold M=r, N=lane.
        const float G00 = __shfl(G[0], 0, 32);
        const float G01 = __shfl(G[0], 1, 32);
        const float G02 = __shfl(G[0], 2, 32);
        const float G11 = __shfl(G[1], 1, 32);
        const float G12 = __shfl(G[1], 2, 32);
        const float G22 = __shfl(G[2], 2, 32);

        // ---- symmetric 3x3 eigenvalues (trig / Smith method) -------------
        const float p1 = G01*G01 + G02*G02 + G12*G12;
        const float q  = (G00 + G11 + G22) * (1.0f / 3.0f);
        const float d0 = G00 - q, d1 = G11 - q, d2 = G22 - q;
        const float p2 = d0*d0 + d1*d1 + d2*d2 + 2.0f * p1;
        float e1, e2, e3;
        if (p2 <= 1e-30f) {
            e1 = e2 = e3 = q;
        } else {
            const float p    = sqrtf(p2 * (1.0f / 6.0f));
            const float ip   = 1.0f / p;
            const float B00 = d0*ip, B11 = d1*ip, B22 = d2*ip;
            const float B01 = G01*ip, B02 = G02*ip, B12 = G12*ip;
            float r = 0.5f * (B00 * (B11*B22 - B12*B12)
                            - B01 * (B01*B22 - B12*B02)
                            + B02 * (B01*B12 - B11*B02));
            r = fminf(1.0f, fmaxf(-1.0f, r));
            const float phi = acosf(r) * (1.0f / 3.0f);
            e1 = q + 2.0f * p * cosf(phi);
            e3 = q + 2.0f * p * cosf(phi + 2.0943951023931953f); // +2π/3
            e2 = 3.0f * q - e1 - e3;
        }
        const float s0 = sqrtf(fmaxf(e1, 0.0f));
        const float s1 = sqrtf(fmaxf(e2, 0.0f));
        const float s2 = sqrtf(fmaxf(e3, 0.0f));
        const float dsgn = (detC > 0.0f) ? 1.0f : ((detC < 0.0f) ? -1.0f : 0.0f);
        const float traceSR = s0 + s1 + dsgn * s2;
        const float rmsd = sqrtf(fmaxf(E - 2.0f * traceSR, 0.0f) * inv_n);

        if (lane == 0) out[b] = rmsd;
    }
}

extern "C" void kernel_launch(void* const* d_in, const int* in_sizes, int n_in,
                              void* d_out, int out_size, void* d_ws, size_t ws_size,
                              hipStream_t stream) {
    const float* x  = (const float*)d_in[0];
    const float* y  = (const float*)d_in[1];
    const int*   na = (const int*)d_in[2];
    float* out = (float*)d_out;

    const int B = in_sizes[2];
    const int N = in_sizes[0] / (3 * B);

    kabsch_rmsd_kernel<<<B, THREADS, 0, stream>>>(x, y, na, out, N);
}